// CorrBlock_15951508538008
// MI455X (gfx1250) — compile-verified
//
#include <hip/hip_runtime.h>
#include <hip/hip_bf16.h>

// Problem constants (from the reference)
#define BATCH   2
#define CDIM    256
#define HH      64
#define WW      128
#define HW      8192        // HH*WW
#define NUM_LVL 4
#define RADIUS  4
#define SIDE    9
#define K2      81          // SIDE*SIDE
#define MCHUNK  2048        // M tile processed per GEMM+sample pass

typedef float v2f __attribute__((ext_vector_type(2)));
typedef float v8f __attribute__((ext_vector_type(8)));

// ---------------------------------------------------------------------------
// 2x2 average pooling over the spatial dims of an (n_ch, Hs, Ws) tensor.
// Used on fmap2: pool_l(corr) == f1^T * pool_l(f2) / sqrt(C)  (dot is linear).
// ---------------------------------------------------------------------------
__global__ void corrblk_pool2x2(const float* __restrict__ src,
                                float* __restrict__ dst,
                                int n_ch, int Hs, int Ws) {
    const int Hd = Hs >> 1, Wd = Ws >> 1;
    const int total = n_ch * Hd * Wd;
    int tid = blockIdx.x * blockDim.x + threadIdx.x;
    if (tid >= total) return;
    const int j = tid % Wd;
    int t = tid / Wd;
    const int i = t % Hd;
    const int c = t / Hd;
    const float* s = src + ((size_t)c * Hs + 2 * i) * Ws + 2 * j;
    dst[tid] = 0.25f * (s[0] + s[1] + s[Ws] + s[Ws + 1]);
}

// ---------------------------------------------------------------------------
// corr_chunk[m_local, n] = (1/16) * sum_k f1[k, m0+m_local] * f2l[k, n]
//   f1  : (CDIM, HW)  batch slice of fmap1  (A^T layout -> coalesced A loads)
//   f2l : (CDIM, Nl)  pooled fmap2 level    (B panel, staged through LDS)
// Block = 256 threads = 8 waves. Each wave computes a 64(M) x 32(N) tile:
// eight 16x16 V_WMMA_F32_16X16X4_F32 accumulators; the two N-subtiles reuse
// the same four A fragments (halves VMEM instructions per WMMA vs 64x16).
// B panel is kept as two 64x16 LDS halves (row stride 16 -> conflict-free,
// and the k/k+1 pair fuses into ds_load_2addr_b32).
// ---------------------------------------------------------------------------
__global__ void __launch_bounds__(256)
corrblk_gemm(const float* __restrict__ f1,
             const float* __restrict__ f2l,
             float* __restrict__ corr,
             int m0, int Nl) {
    __shared__ float ldsB[2][64 * 16];    // two 64K x 16N halves, 8 KB total

    const int t     = threadIdx.x;
    const int lane  = t & 31;
    const int wave  = t >> 5;
    const int nbase = blockIdx.x * 32;
    const int mblk  = blockIdx.y * 512 + wave * 64;   // local M base (chunk coords)
    const int mrow  = lane & 15;                       // A: M = lane%16
    const int khalf = (lane >> 4) << 1;                // A/B: lanes 16-31 hold K+2

    v8f acc[4][2];
#pragma unroll
    for (int s = 0; s < 4; ++s)
#pragma unroll
        for (int nb = 0; nb < 2; ++nb)
            acc[s][nb] = (v8f){0.f,0.f,0.f,0.f,0.f,0.f,0.f,0.f};

    const int bn = t & 31;          // LDS fill: column 0..31
    const int bk = t >> 5;          // LDS fill: row 0..7 (x8 passes)
    const int bh = bn >> 4;         // which half
    const int bc = bn & 15;         // column within half

    for (int kc = 0; kc < CDIM; kc += 64) {
#pragma unroll
        for (int p = 0; p < 8; ++p) {
            ldsB[bh][(bk + p * 8) * 16 + bc] =
                f2l[(size_t)(kc + bk + p * 8) * Nl + nbase + bn];
        }
        __syncthreads();

#pragma unroll
        for (int k4 = 0; k4 < 64; k4 += 4) {
            const int kk = k4 + khalf;
            v2f bfrag[2];
#pragma unroll
            for (int nb = 0; nb < 2; ++nb) {
                bfrag[nb].x = ldsB[nb][kk * 16 + (lane & 15)];
                bfrag[nb].y = ldsB[nb][(kk + 1) * 16 + (lane & 15)];
            }
            const float* aptr = f1 + (size_t)(kc + kk) * HW + (m0 + mblk + mrow);
#pragma unroll
            for (int s = 0; s < 4; ++s) {
                v2f afrag;
                afrag.x = aptr[s * 16];        // (k,   m)
                afrag.y = aptr[s * 16 + HW];   // (k+1, m)
#pragma unroll
                for (int nb = 0; nb < 2; ++nb) {
                    acc[s][nb] = __builtin_amdgcn_wmma_f32_16x16x4_f32(
                        false, afrag, false, bfrag[nb], (short)0, acc[s][nb],
                        false, false);
                }
            }
        }
        __syncthreads();
    }

    // C/D layout: VGPR r -> M=r (lanes 0-15), M=8+r (lanes 16-31); N = lane%16
    const int ncol   = lane & 15;
    const int rowoff = (lane >> 4) << 3;
#pragma unroll
    for (int s = 0; s < 4; ++s) {
#pragma unroll
        for (int r = 0; r < 8; ++r) {
            const int m_local = mblk + s * 16 + rowoff + r;
            float* dst = corr + (size_t)m_local * Nl + nbase + ncol;
            dst[0]  = acc[s][0][r] * 0.0625f;
            dst[16] = acc[s][1][r] * 0.0625f;
        }
    }
}

// ---------------------------------------------------------------------------
// Bilinear lookup of the 9x9 neighborhood at one pyramid level for one
// M-chunk of queries. Consecutive threads = consecutive queries -> coalesced
// coord reads and output stores; gathers hit the chunk still resident in L2.
// ---------------------------------------------------------------------------
__global__ void corrblk_sample(const float* __restrict__ corr,
                               const float* __restrict__ cc,
                               float* __restrict__ out,
                               int b, int lvl, int m0, int Hl, int Wl) {
    int tid = blockIdx.x * blockDim.x + threadIdx.x;
    if (tid >= MCHUNK * K2) return;
    const int m_local = tid & (MCHUNK - 1);
    const int kk      = tid >> 11;              // MCHUNK == 2048
    const int m = m0 + m_local;
    const int i = m >> 7;                        // / WW
    const int j = m & (WW - 1);

    const float x = cc[((size_t)(b * 2 + 0) * HH + i) * WW + j];
    const float y = cc[((size_t)(b * 2 + 1) * HH + i) * WW + j];
    const float inv = 1.0f / (float)(1 << lvl);
    const float di = (float)(kk / SIDE - RADIUS);
    const float dj = (float)(kk % SIDE - RADIUS);
    const float sx = x * inv + di;
    const float sy = y * inv + dj;

    const float x0 = floorf(sx), y0 = floorf(sy);
    const float wx = sx - x0,    wy = sy - y0;
    const float wmax = (float)(Wl - 1), hmax = (float)(Hl - 1);
    const float* img = corr + (size_t)m_local * (Hl * Wl);

    auto gather = [&](float xi, float yi) -> float {
        const bool valid = (xi >= 0.f) && (xi <= wmax) && (yi >= 0.f) && (yi <= hmax);
        const int xc = (int)fminf(fmaxf(xi, 0.f), wmax);
        const int yc = (int)fminf(fmaxf(yi, 0.f), hmax);
        const float v = img[yc * Wl + xc];
        return valid ? v : 0.f;
    };

    const float v = gather(x0,       y0      ) * (1.f - wx) * (1.f - wy)
                  + gather(x0 + 1.f, y0      ) * wx          * (1.f - wy)
                  + gather(x0,       y0 + 1.f) * (1.f - wx) * wy
                  + gather(x0 + 1.f, y0 + 1.f) * wx          * wy;

    out[(((size_t)b * (NUM_LVL * K2) + lvl * K2 + kk) * HH + i) * WW + j] = v;
}

// ---------------------------------------------------------------------------
extern "C" void kernel_launch(void* const* d_in, const int* in_sizes, int n_in,
                              void* d_out, int out_size, void* d_ws, size_t ws_size,
                              hipStream_t stream) {
    const float* f1 = (const float*)d_in[0];   // (B, C, H, W)
    const float* f2 = (const float*)d_in[1];   // (B, C, H, W)
    const float* cc = (const float*)d_in[2];   // (B, 2, H, W)
    float* out = (float*)d_out;                // (B, 324, H, W)

    // Workspace layout:
    //   [0, 64 MiB)        : corr chunk (MCHUNK x 8192 fp32, reused per pass)
    //   then pooled fmap2  : lvl1 (B*C*2048), lvl2 (B*C*512), lvl3 (B*C*128)
    float* corr = (float*)d_ws;
    float* p1 = corr + (size_t)MCHUNK * HW;
    float* p2 = p1 + (size_t)BATCH * CDIM * 2048;
    float* p3 = p2 + (size_t)BATCH * CDIM * 512;

    {   // Build the fmap2 pooling pyramid (cheap: a few MB total)
        int n1 = BATCH * CDIM * 32 * 64;
        corrblk_pool2x2<<<(n1 + 255) / 256, 256, 0, stream>>>(f2, p1, BATCH * CDIM, 64, 128);
        int n2 = BATCH * CDIM * 16 * 32;
        corrblk_pool2x2<<<(n2 + 255) / 256, 256, 0, stream>>>(p1, p2, BATCH * CDIM, 32, 64);
        int n3 = BATCH * CDIM * 8 * 16;
        corrblk_pool2x2<<<(n3 + 255) / 256, 256, 0, stream>>>(p2, p3, BATCH * CDIM, 16, 32);
    }

    for (int b = 0; b < BATCH; ++b) {
        const float* f1b = f1 + (size_t)b * CDIM * HW;
        const float* plv[NUM_LVL] = {
            f2 + (size_t)b * CDIM * HW,
            p1 + (size_t)b * CDIM * 2048,
            p2 + (size_t)b * CDIM * 512,
            p3 + (size_t)b * CDIM * 128
        };
        for (int lvl = 0; lvl < NUM_LVL; ++lvl) {
            const int Hl = HH >> lvl, Wl = WW >> lvl, Nl = Hl * Wl;
            for (int m0 = 0; m0 < HW; m0 += MCHUNK) {
                dim3 grid(Nl / 32, MCHUNK / 512);
                corrblk_gemm<<<grid, 256, 0, stream>>>(f1b, plv[lvl], corr, m0, Nl);
                const int nthr = MCHUNK * K2;
                corrblk_sample<<<(nthr + 255) / 256, 256, 0, stream>>>(
                    corr, cc, out, b, lvl, m0, Hl, Wl);
            }
        }
    }
}